// CNSN_46196668236411
// MI455X (gfx1250) — compile-verified
//
#include <hip/hip_runtime.h>
#include <math.h>

// MI455X single-pass fused channel-norm + tiny-MLP gating.
// Traffic: read 256 MiB (NT), write 256 MiB (NT) -> ~22us @ 23.3 TB/s.
// MLP (2->16->1, x2) done in-kernel with V_WMMA_F32_16X16X4_F32 on wave 0.

typedef float v2f __attribute__((ext_vector_type(2)));
typedef float v4f __attribute__((ext_vector_type(4)));
typedef float v8f __attribute__((ext_vector_type(8)));

#define HW      4096      // 64*64 plane
#define PLANES  8         // channel planes per block (128 KiB LDS stage)
#define TPB     512       // 16 wave32 waves
#define NWAVES  (TPB/32)
#define F4T     ((PLANES*HW/4)/TPB)   // float4 per thread = 16

__global__ __launch_bounds__(TPB) void cnsn_fused(
    const float* __restrict__ x,
    const float* __restrict__ wm1, const float* __restrict__ bm1,
    const float* __restrict__ wm2, const float* __restrict__ bm2,
    const float* __restrict__ ws1, const float* __restrict__ bs1,
    const float* __restrict__ ws2, const float* __restrict__ bs2,
    float* __restrict__ out)
{
    extern __shared__ float sx[];                 // PLANES*HW floats (128 KiB)
    __shared__ float sred[NWAVES][PLANES][2];     // per-wave partial sums
    __shared__ float s_stat[PLANES][2];           // mean, std
    __shared__ float s_coef[PLANES][2];           // scale, bias

    const int t    = threadIdx.x;
    const int wave = t >> 5;
    const int lane = t & 31;
    const long long base = (long long)blockIdx.x * (PLANES * HW);

    const v4f* __restrict__ xg = (const v4f*)(x + base);
    v4f* __restrict__ og = (v4f*)(out + base);
    v4f* sx4 = (v4f*)sx;

    // ---- Phase 1: stream global -> LDS, accumulate per-plane sum / sumsq ----
    float psum[PLANES], psq[PLANES];
#pragma unroll
    for (int p = 0; p < PLANES; ++p) { psum[p] = 0.f; psq[p] = 0.f; }

#pragma unroll
    for (int k = 0; k < F4T; ++k) {
        const int i4 = t + k * TPB;               // element offset = 4t + 2048k
        v4f v = __builtin_nontemporal_load(&xg[i4]);
        sx4[i4] = v;
        const int p = k >> 1;                     // plane id, compile-time per k
        psum[p] += (v[0] + v[1]) + (v[2] + v[3]);
        psq[p]  += (v[0]*v[0] + v[1]*v[1]) + (v[2]*v[2] + v[3]*v[3]);
    }

    // ---- Phase 2: wave32 xor-tree reduce, then cross-wave combine ----
#pragma unroll
    for (int p = 0; p < PLANES; ++p) {
        float s = psum[p], q = psq[p];
#pragma unroll
        for (int m = 16; m >= 1; m >>= 1) {
            s += __shfl_xor(s, m, 32);
            q += __shfl_xor(q, m, 32);
        }
        if (lane == 0) { sred[wave][p][0] = s; sred[wave][p][1] = q; }
    }
    __syncthreads();

    if (t < PLANES) {
        float s = 0.f, q = 0.f;
#pragma unroll
        for (int w = 0; w < NWAVES; ++w) { s += sred[w][t][0]; q += sred[w][t][1]; }
        const float mean = s * (1.0f / HW);
        const float var  = (q - s * mean) * (1.0f / (HW - 1));   // ddof=1
        s_stat[t][0] = mean;
        s_stat[t][1] = sqrtf(var + 1e-5f);
    }
    __syncthreads();

    // ---- Phase 3: both 2->16->1 MLPs on wave 0 via WMMA f32 16x16x4 ----
    if (wave == 0) {
        // A (16x4): lanes 0-15 hold K=0,1 (mean,std); lanes 16-31 hold K=2,3 (=0)
        v2f a; a[0] = 0.f; a[1] = 0.f;
        if (lane < PLANES) { a[0] = s_stat[lane][0]; a[1] = s_stat[lane][1]; }
        const int  n  = lane & 15;
        const bool lo = lane < 16;
        // B (4x16) = w1^T: VGPR0 = row K0 (lanes 0-15) / K2 (16-31); VGPR1 = K1/K3
        v2f b_m, b_s;
        b_m[0] = lo ? wm1[2*n + 0] : 0.f;  b_m[1] = lo ? wm1[2*n + 1] : 0.f;
        b_s[0] = lo ? ws1[2*n + 0] : 0.f;  b_s[1] = lo ? ws1[2*n + 1] : 0.f;
        // C preloaded with layer-1 bias (per column N = lane&15)
        v8f c_m, c_s;
        const float b1m = bm1[n], b1s = bs1[n];
#pragma unroll
        for (int v = 0; v < 8; ++v) { c_m[v] = b1m; c_s[v] = b1s; }

        c_m = __builtin_amdgcn_wmma_f32_16x16x4_f32(false, a, false, b_m,
                                                    (short)0, c_m, false, false);
        c_s = __builtin_amdgcn_wmma_f32_16x16x4_f32(false, a, false, b_s,
                                                    (short)0, c_s, false, false);

        // Layer 2: relu, multiply by w2[N], reduce across the 16-lane group.
        const float w2m = wm2[n], w2s = ws2[n];
        const float b2m = bm2[0], b2s = bs2[0];
        float meanw[8], stdw[8];
#pragma unroll
        for (int v = 0; v < 8; ++v) {   // VGPR v <-> row M = v (lanes 0-15)
            float hm = fmaxf(c_m[v], 0.f) * w2m;
            float hs = fmaxf(c_s[v], 0.f) * w2s;
#pragma unroll
            for (int m = 1; m < 16; m <<= 1) {
                hm += __shfl_xor(hm, m, 32);
                hs += __shfl_xor(hs, m, 32);
            }
            meanw[v] = 1.f / (1.f + expf(-(hm + b2m)));
            stdw[v]  = 1.f / (1.f + expf(-(hs + b2s)));
        }
        if (lane == 0) {
            // out = x*std_w + mean*(mean_w - std_w)
#pragma unroll
            for (int v = 0; v < PLANES; ++v) {
                const float mw = meanw[v], sw = stdw[v];
                s_coef[v][0] = sw;
                s_coef[v][1] = s_stat[v][0] * (mw - sw);
            }
        }
    }
    __syncthreads();

    // ---- Phase 4: apply affine from LDS, NT store ----
#pragma unroll
    for (int k = 0; k < F4T; ++k) {
        const int i4 = t + k * TPB;
        const int p  = k >> 1;
        const float sc = s_coef[p][0], bi = s_coef[p][1];
        v4f v = sx4[i4];
        v4f o;
        o[0] = v[0]*sc + bi; o[1] = v[1]*sc + bi;
        o[2] = v[2]*sc + bi; o[3] = v[3]*sc + bi;
        __builtin_nontemporal_store(o, &og[i4]);
    }
}

extern "C" void kernel_launch(void* const* d_in, const int* in_sizes, int n_in,
                              void* d_out, int out_size, void* d_ws, size_t ws_size,
                              hipStream_t stream) {
    (void)n_in; (void)out_size; (void)d_ws; (void)ws_size;
    const float* x   = (const float*)d_in[0];
    const float* wm1 = (const float*)d_in[1];
    const float* bm1 = (const float*)d_in[2];
    const float* wm2 = (const float*)d_in[3];
    const float* bm2 = (const float*)d_in[4];
    const float* ws1 = (const float*)d_in[5];
    const float* bs1 = (const float*)d_in[6];
    const float* ws2 = (const float*)d_in[7];
    const float* bs2 = (const float*)d_in[8];
    float* out = (float*)d_out;

    const int planes = in_sizes[0] / HW;          // B*C = 16384
    const int grid   = planes / PLANES;           // 2048 blocks
    const size_t shmem = (size_t)PLANES * HW * sizeof(float);  // 128 KiB

    hipFuncSetAttribute((const void*)cnsn_fused,
                        hipFuncAttributeMaxDynamicSharedMemorySize, (int)shmem);
    cnsn_fused<<<grid, TPB, shmem, stream>>>(x, wm1, bm1, wm2, bm2,
                                             ws1, bs1, ws2, bs2, out);
}